// CascadeMatching_26877905338833
// MI455X (gfx1250) — compile-verified
//
#include <hip/hip_runtime.h>
#include <hip/hip_bf16.h>
#include <float.h>

typedef __attribute__((ext_vector_type(2))) float v2f;
typedef __attribute__((ext_vector_type(8))) float v8f;

#define C_DIM 64
#define K_DIM 100
#define NTILE 7                 // ceil(100/16)
#define SIM_SCALE 0.15625f      // (1/sqrt(C))^2 / TEMPERATURE = 1/(64*0.1)
#define NEG_INF_REF (-1.0e9f)   // reference's -INF

// One wave32 per query. WMMA scheme: A[i,c] = q[c] (query replicated over the
// 16 M-rows), B[c,j] = key_j[c] (16 gathered keys as columns). D[i,j] = q.key_j,
// so lane j holds sim of key j in VGPR0 of the accumulator -> no shuffle
// reduction for the dot products. 16 chunk-WMMAs (K=4) cover C=64.
__global__ __launch_bounds__(256)
void cascade_match_wmma(const float* __restrict__ fq,
                        const float* __restrict__ fk,
                        const int*   __restrict__ idx,
                        const int*   __restrict__ mq,
                        const int*   __restrict__ mk,
                        float* __restrict__ conf,
                        float* __restrict__ nconf,
                        int*   __restrict__ nidx,
                        int nq, int hw)
{
    const int lane = (int)(threadIdx.x & 31u);
    const int wave = (int)((blockIdx.x * blockDim.x + threadIdx.x) >> 5);
    if (wave >= nq) return;                 // wave-uniform; EXEC stays all-1s
    const int b   = wave / hw;
    const int hi  = lane >> 4;              // 0 = lanes 0-15, 1 = lanes 16-31
    const int col = lane & 15;              // key column within a 16-key tile

    // ---- query row: lane holds q[2*lane], q[2*lane+1] -----------------------
    const float* qrow = fq + (size_t)wave * C_DIM;
    float2 qp = *(const float2*)(qrow + 2 * lane);

    // ---- build replicated-query A operand layout ----------------------------
    // chunk u covers channels 4u..4u+3:
    //   lanes 0-15 need (q[4u], q[4u+1]) = lane(2u).{x,y}
    //   lanes 16-31 need (q[4u+2], q[4u+3]) = lane(2u+1).{x,y}
    float a0[16], a1[16];
#pragma unroll
    for (int u = 0; u < 16; ++u) {
        int src = 2 * u + hi;
        a0[u] = __shfl(qp.x, src, 32);
        a1[u] = __shfl(qp.y, src, 32);
    }

    const int    mqv      = mq[wave];
    const size_t idx_base = (size_t)wave * K_DIM;

    float sims[NTILE];
#pragma unroll
    for (int t = 0; t < NTILE; ++t) {
        int k     = 16 * t + col;
        int valid = (k < K_DIM);
        int kk    = valid ? k : (K_DIM - 1);      // clamp: safe load for pads
        int iv    = idx[idx_base + kk];           // lanes j and j+16 load same
        int wm    = mk[b * hw + iv] & mqv;        // window mask (0/1 values)
        const float* krow = fk + ((size_t)(b * hw) + iv) * C_DIM;

        v8f acc = {};
#pragma unroll
        for (int u = 0; u < 16; ++u) {
            // B operand: lane j (low half) = key_j[4u..4u+1],
            //            lane j+16        = key_j[4u+2..4u+3]
            float2 bp = *(const float2*)(krow + 4 * u + 2 * hi);
            v2f av = { a0[u], a1[u] };
            v2f bv = { bp.x, bp.y };
            acc = __builtin_amdgcn_wmma_f32_16x16x4_f32(
                false, av, false, bv, (short)0, acc, false, false);
        }
        float dot = acc[0];                       // (M=0, N=col) = q . key_col
        sims[t] = !valid ? -FLT_MAX
                         : (wm ? dot * SIM_SCALE : NEG_INF_REF);
    }

    // ---- softmax over K (values live at k = 16t + col; halves duplicate) ----
    float m = sims[0];
#pragma unroll
    for (int t = 1; t < NTILE; ++t) m = fmaxf(m, sims[t]);
#pragma unroll
    for (int off = 1; off < 32; off <<= 1)
        m = fmaxf(m, __shfl_xor(m, off, 32));     // max is dup-safe

    float e[NTILE];
    float sloc = 0.0f;
#pragma unroll
    for (int t = 0; t < NTILE; ++t) {
        int k = 16 * t + col;
        e[t] = (k < K_DIM) ? __expf(sims[t] - m) : 0.0f;
        sloc += e[t];
    }
    float s = hi ? 0.0f : sloc;                   // de-dup upper half for sum
#pragma unroll
    for (int off = 1; off < 32; off <<= 1)
        s += __shfl_xor(s, off, 32);

    // argmax: max exp is exactly 1.0 (sims==m); first occurrence = min k
    int kbest = 0x7fffffff;
#pragma unroll
    for (int t = 0; t < NTILE; ++t) {
        int k = 16 * t + col;
        if (k < K_DIM && sims[t] == m) kbest = min(kbest, k);
    }
#pragma unroll
    for (int off = 1; off < 32; off <<= 1)
        kbest = min(kbest, __shfl_xor(kbest, off, 32));

    // ---- outputs ------------------------------------------------------------
    float invs = 1.0f / s;
    const size_t cbase = (size_t)wave * K_DIM;
    if (!hi) {
#pragma unroll
        for (int t = 0; t < NTILE; ++t) {
            int k = 16 * t + col;
            if (k < K_DIM) conf[cbase + k] = e[t] * invs;   // coalesced 64B/t
        }
        if (lane == 0) {
            nconf[wave] = invs;                   // max conf = 1.0 * (1/s)
            nidx[wave]  = idx[idx_base + kbest];  // uniform kbest: 1 reload
        }
    }
}

extern "C" void kernel_launch(void* const* d_in, const int* in_sizes, int n_in,
                              void* d_out, int out_size, void* d_ws, size_t ws_size,
                              hipStream_t stream)
{
    const float* f0  = (const float*)d_in[0];   // feat_c0 [B,HW,C] f32
    const float* f1  = (const float*)d_in[1];   // feat_c1 [B,HW,C] f32
    const int*  i01  = (const int*)d_in[2];     // idx_c01 [B,HW,K] i32
    const int*  i10  = (const int*)d_in[3];     // idx_c10 [B,HW,K] i32
    const int*  m0   = (const int*)d_in[4];     // mask_c0 [B,HW]   i32
    const int*  m1   = (const int*)d_in[5];     // mask_c1 [B,HW]   i32

    const int nq = in_sizes[4];                 // B*HW = 38400
    const int hw = 19200;                       // fixed by reference shapes

    // d_out layout (flat, return order):
    // conf01[nq*K] | nconf01[nq] | nidx01[nq] | conf10[nq*K] | nconf10[nq] | nidx10[nq]
    float* base = (float*)d_out;
    const size_t ck = (size_t)nq * K_DIM;
    float* conf01  = base;
    float* nconf01 = base + ck;
    int*   nidx01  = (int*)(base + ck + nq);
    float* conf10  = base + ck + 2 * (size_t)nq;
    float* nconf10 = conf10 + ck;
    int*   nidx10  = (int*)(conf10 + ck + nq);

    const int threads = 256;                    // 8 wave32 per block
    const int blocks  = (nq + 7) / 8;           // one wave per query (exact)

    cascade_match_wmma<<<blocks, threads, 0, stream>>>(
        f0, f1, i01, m0, m1, conf01, nconf01, nidx01, nq, hw);
    cascade_match_wmma<<<blocks, threads, 0, stream>>>(
        f1, f0, i10, m1, m0, conf10, nconf10, nidx10, nq, hw);
}